// yolov2_loss_23252952941261
// MI455X (gfx1250) — compile-verified
//
#include <hip/hip_runtime.h>
#include <hip/hip_bf16.h>

#define NUM_CLASSES 80
#define NUM_ANCHORS 5
#define MAX_BOXES   50
#define NB 64
#define NH 32
#define NW 32
#define PLANE (NH*NW)                       /* 1024 */
#define GRID  (NB*NUM_ANCHORS*NH*NW)        /* 327680 */
#define NCH   (5+NUM_CLASSES)               /* 85 */

typedef __attribute__((ext_vector_type(2))) float v2f;
typedef __attribute__((ext_vector_type(8))) float v8f;

__device__ __constant__ float c_aw[5] = {42.0f/32, 102.0f/32, 161.0f/32, 303.0f/32, 359.0f/32};
__device__ __constant__ float c_ah[5] = {55.0f/32, 128.0f/32, 259.0f/32, 155.0f/32, 320.0f/32};

__device__ __forceinline__ float softplus_f(float x) {
    // log1p(exp(x)) stable
    return fmaxf(x, 0.0f) + log1pf(__expf(-fabsf(x)));
}
// -(t*clip(log sigmoid(o),-100) + (1-t)*clip(log1p(-sigmoid(o)),-100))
__device__ __forceinline__ float bce_logit(float o, float t) {
    float lp = fminf(softplus_f(-o), 100.0f);
    float ln = fminf(softplus_f(o),  100.0f);
    return t * lp + (1.0f - t) * ln;
}

// ---------------- kernel 0: zero scratch regions we accumulate into ----------
__global__ void zero_ws(float* objclr /*2*GRID floats*/, unsigned* mask /*3*GRID*/,
                        int* count, float* acc) {
    size_t idx = (size_t)blockIdx.x * 256 + threadIdx.x;
    if (idx < 2u * GRID) objclr[idx] = 0.0f;
    if (idx < 3u * GRID) mask[idx] = 0u;
    if (idx == 0) { *count = 0; *acc = 0.0f; }
}

// ---------------- kernel 1: build_targets (tiny) -----------------------------
__global__ void build_targets(const float* __restrict__ target,
                              float* obj, float* clr, float* tx, float* ty,
                              float* tw, float* th, unsigned* mask,
                              int* list, int* count) {
    int b = blockIdx.x;
    int t = threadIdx.x;
    __shared__ int nz[MAX_BOXES];
    const float* tg = target + (size_t)b * MAX_BOXES * 5;
    if (t < MAX_BOXES) nz[t] = (tg[t*5 + 1] != 0.0f) ? 1 : 0;
    __syncthreads();
    if (t >= MAX_BOXES) return;
    bool valid = true;
    for (int k = 0; k <= t; ++k) valid = valid && (nz[k] != 0);   // cumprod
    if (!valid) return;

    float cls = tg[t*5+0], x = tg[t*5+1], y = tg[t*5+2], w = tg[t*5+3], h = tg[t*5+4];
    float gx = x * NW, gy = y * NH, gw = w * NW, gh = h * NH;
    int gi = min(max((int)gx, 0), NW-1);
    int gj = min(max((int)gy, 0), NH-1);

    float ious[NUM_ANCHORS];
    float best_iou = -1.0f; int best = 0;
    #pragma unroll
    for (int a = 0; a < NUM_ANCHORS; ++a) {
        float aw = c_aw[a], ah = c_ah[a];
        float inter = fmaxf(fminf(gw, aw) + 1.0f, 0.0f) * fmaxf(fminf(gh, ah) + 1.0f, 0.0f);
        float uni = (gw+1.0f)*(gh+1.0f) + (aw+1.0f)*(ah+1.0f) - inter + 1e-16f;
        ious[a] = inter / uni;
        if (ious[a] > best_iou) { best_iou = ious[a]; best = a; }   // first-max (argmax)
    }
    #pragma unroll
    for (int a = 0; a < NUM_ANCHORS; ++a) {
        if (ious[a] > 0.6f || a == best) {
            int c = ((b*NUM_ANCHORS + a)*NH + gj)*NW + gi;
            clr[c] = 1.0f;                // all racers write 1.0 -> benign
        }
    }
    int cell = ((b*NUM_ANCHORS + best)*NH + gj)*NW + gi;
    unsigned old = atomicCAS((unsigned*)&obj[cell], 0u, __float_as_uint(1.0f));
    if (old == 0u) {                       // dedup: each obj cell listed once
        int pos = atomicAdd(count, 1);
        list[pos] = cell;
    }
    tx[cell] = gx - (float)gi;
    ty[cell] = gy - (float)gj;
    tw[cell] = logf(fmaxf(gw, 1e-8f) / c_aw[best]);
    th[cell] = logf(fmaxf(gh, 1e-8f) / c_ah[best]);
    int k = min(max((int)cls, 0), NUM_CLASSES-1);
    atomicOr(&mask[(size_t)cell*3 + (k >> 5)], 1u << (k & 31));
}

// ---------------- kernel 2: noobj conf term over all cells -------------------
// One block per (b,anchor): async-stage the 4KB conf-logit plane and the 4KB
// clear plane into LDS, compute, then WMMA-reduce 256 partials.
__global__ void noobj_pass(const float* __restrict__ out,
                           const float* __restrict__ clr, float* acc) {
    __shared__ __align__(16) float sconf[PLANE];
    __shared__ __align__(16) float sclr[PLANE];
    __shared__ float red[256];
    int tid = threadIdx.x;
    int bi  = blockIdx.x;                  // b*5 + a
    int b = bi / NUM_ANCHORS, a = bi % NUM_ANCHORS;

    const float* gp = out + ((size_t)(b*(NUM_ANCHORS*NCH) + a*NCH + 4)) * PLANE + tid*4;
    const float* cp = clr + (size_t)bi * PLANE + tid*4;
    unsigned l0 = (unsigned)(size_t)(&sconf[tid*4]);   // low 32b of LDS-aperture addr = LDS offset
    unsigned l1 = (unsigned)(size_t)(&sclr[tid*4]);
    asm volatile("global_load_async_to_lds_b128 %0, %1, off" :: "v"(l0), "v"(gp) : "memory");
    asm volatile("global_load_async_to_lds_b128 %0, %1, off" :: "v"(l1), "v"(cp) : "memory");
    asm volatile("s_wait_asynccnt 0x0" ::: "memory");
    __syncthreads();

    float partial = 0.0f;
    #pragma unroll
    for (int k = 0; k < 4; ++k) {
        int u = tid + k*256;
        float o  = sconf[u];
        float nb = 1.0f - sclr[u];
        partial += nb * 0.5f * fminf(softplus_f(o), 100.0f);
    }
    red[tid] = partial;
    __syncthreads();

    if (tid < 32) {                         // wave 0, EXEC all-ones for WMMA
        v8f accv = {0.f,0.f,0.f,0.f,0.f,0.f,0.f,0.f};
        v2f ones = {1.0f, 1.0f};
        #pragma unroll
        for (int c = 0; c < 4; ++c) {       // 4 x (64 values) -> row-sums in D
            v2f av = {red[c*64 + tid], red[c*64 + 32 + tid]};
            accv = __builtin_amdgcn_wmma_f32_16x16x4_f32(
                false, av, false, ones, (short)0, accv, false, false);
        }
        float s = accv[0]+accv[1]+accv[2]+accv[3]+accv[4]+accv[5]+accv[6]+accv[7];
        s += __shfl_xor(s, 16, 32);         // combine lane halves
        if (tid == 0) atomicAdd(acc, s);
    }
}

// ---------------- kernel 3: obj cells (<=3200), one wave per cell ------------
__global__ void obj_pass(const float* __restrict__ out,
                         const float* tx, const float* ty,
                         const float* tw, const float* th,
                         const unsigned* mask, const int* list,
                         const int* count, float* acc) {
    int wave = (int)((blockIdx.x * blockDim.x + threadIdx.x) >> 5);
    int lane = threadIdx.x & 31;
    if (wave >= *count) return;
    int cell = list[wave];
    int i = cell & (NW-1);
    int j = (cell >> 5) & (NH-1);
    int ba = cell >> 10;                    // b*5 + a
    int b = ba / NUM_ANCHORS, a = ba % NUM_ANCHORS;
    size_t base = ((size_t)(b*(NUM_ANCHORS*NCH) + a*NCH)) * PLANE + (size_t)j*NW + i;

    float txv = tx[cell], tyv = ty[cell], twv = tw[cell], thv = th[cell];
    float s = 0.0f;
    for (int c = lane; c < NCH; c += 32) {
        float o = out[base + (size_t)c * PLANE];
        if (c == 0)      s += bce_logit(o, txv);
        else if (c == 1) s += bce_logit(o, tyv);
        else if (c == 2) { float d = o - twv; s += d*d; }
        else if (c == 3) { float d = o - thv; s += d*d; }
        else if (c == 4) s += fminf(softplus_f(-o), 100.0f);  // bce(pred_conf,1)
        else {
            int k = c - 5;
            float t = ((mask[(size_t)cell*3 + (k >> 5)] >> (k & 31)) & 1u) ? 1.0f : 0.0f;
            s += bce_logit(o, t);
        }
    }
    #pragma unroll
    for (int off = 16; off; off >>= 1) s += __shfl_xor(s, off, 32);
    if (lane == 0) atomicAdd(acc, s);
}

// ---------------- kernel 4: finalize ----------------------------------------
__global__ void finalize(const float* acc, float* out) {
    if (threadIdx.x == 0 && blockIdx.x == 0) out[0] = acc[0] / (float)NB;
}

extern "C" void kernel_launch(void* const* d_in, const int* in_sizes, int n_in,
                              void* d_out, int out_size, void* d_ws, size_t ws_size,
                              hipStream_t stream) {
    const float* output = (const float*)d_in[0];
    const float* target = (const float*)d_in[1];
    float* ws   = (float*)d_ws;
    float* obj  = ws;
    float* clr  = ws + 1*(size_t)GRID;
    float* tx   = ws + 2*(size_t)GRID;
    float* ty   = ws + 3*(size_t)GRID;
    float* tw   = ws + 4*(size_t)GRID;
    float* th   = ws + 5*(size_t)GRID;
    unsigned* mask = (unsigned*)(ws + 6*(size_t)GRID);   // 3*GRID u32
    int*   list  = (int*)(ws + 9*(size_t)GRID);          // 4096 ints
    int*   count = (int*)(ws + 9*(size_t)GRID + 4096);
    float* acc   = ws + 9*(size_t)GRID + 4097;

    zero_ws<<<(3*GRID + 255)/256, 256, 0, stream>>>(obj, mask, count, acc);
    build_targets<<<NB, 64, 0, stream>>>(target, obj, clr, tx, ty, tw, th, mask, list, count);
    noobj_pass<<<NB*NUM_ANCHORS, 256, 0, stream>>>(output, clr, acc);
    obj_pass<<<(NB*MAX_BOXES + 7)/8, 256, 0, stream>>>(output, tx, ty, tw, th, mask, list, count, acc);
    finalize<<<1, 32, 0, stream>>>(acc, (float*)d_out);
}